// SpikeFP32OuterProduct_11450382811497
// MI455X (gfx1250) — compile-verified
//
#include <hip/hip_runtime.h>
#include <stdint.h>

typedef __attribute__((ext_vector_type(2))) float v2f;
typedef __attribute__((ext_vector_type(4))) float v4f;
typedef __attribute__((ext_vector_type(8))) float v8f;

#define BATCH 512
#define KDIM  1024
#define NDIM  1024
#define NT    2   // 16x16 N-tiles per wave -> 16x32 strip of C per wave

// ---------------------------------------------------------------------------
// Kernel 1: decode pulse bits -> fp32 values.
//   A [512,1024,32] -> AvT [1024,512]  (transposed so GEMM A-loads are
//                                       contiguous along the WMMA K axis)
//   B [512,1024,32] -> Bv  [512,1024]  (row-major, already B-operand layout)
// Bit extraction: pulses are exactly 0.0f / 1.0f; bit 23 of the IEEE pattern
// distinguishes them (1.0f = 0x3F800000), so no float compares are needed.
// ---------------------------------------------------------------------------
__global__ void __launch_bounds__(256)
decode_kernel(const float* __restrict__ A, const float* __restrict__ B,
              float* __restrict__ AvT, float* __restrict__ Bv) {
    const int idx = blockIdx.x * blockDim.x + threadIdx.x;   // 0 .. 2*512*1024-1
    const bool isB = idx >= BATCH * KDIM;
    const int i = isB ? (idx - BATCH * KDIM) : idx;
    const float* __restrict__ src = (isB ? B : A) + (size_t)i * 32;

    uint32_t u = 0;
#pragma unroll
    for (int j = 0; j < 8; ++j) {
        v4f v = *(const v4f*)(src + j * 4);                  // global_load_b128
        u = (u << 1) | ((__float_as_uint(v.x) >> 23) & 1u);  // MSB-first
        u = (u << 1) | ((__float_as_uint(v.y) >> 23) & 1u);
        u = (u << 1) | ((__float_as_uint(v.z) >> 23) & 1u);
        u = (u << 1) | ((__float_as_uint(v.w) >> 23) & 1u);
    }
    const float val = __uint_as_float(u);

    if (isB) {
        Bv[i] = val;                                         // [b, n]
    } else {
        const int b = i / KDIM, k = i % KDIM;
        AvT[(size_t)k * BATCH + b] = val;                    // [k, b]
    }
}

// ---------------------------------------------------------------------------
// Kernel 2: fp32 WMMA GEMM (C = AvT @ Bv) with fused bit-pulse encode.
// Each wave computes a 16x32 strip of C via V_WMMA_F32_16X16X4_F32.
//
// 16x4 f32 A-operand layout (ISA 7.12.2): lanes 0-15 and 16-31 both cover
// M=0..15; VGPR0 holds K=0 (lanes 0-15) / K=2 (lanes 16-31), VGPR1 holds
// K=1 / K=3. B (4x16) mirrors this with N across lanes.
// C/D: VGPR r -> M=r (lanes 0-15) / M=8+r (lanes 16-31), N = lane & 15.
// ---------------------------------------------------------------------------
__global__ void __launch_bounds__(256)
gemm_encode_kernel(const float* __restrict__ AvT, const float* __restrict__ Bv,
                   float* __restrict__ out) {
    const int wave = (blockIdx.x * blockDim.x + threadIdx.x) >> 5;
    const int lane = threadIdx.x & 31;
    const int half = lane >> 4;      // 0: K pair {0,1}, 1: K pair {2,3}
    const int l15  = lane & 15;

    const int tileM = wave >> 5;                 // 0..63
    const int n0    = (wave & 31) * (16 * NT);   // 0..1023 step 32

    const float* __restrict__ arow = AvT + (size_t)(tileM * 16 + l15) * BATCH;

    v8f acc[NT];
#pragma unroll
    for (int t = 0; t < NT; ++t) acc[t] = (v8f)(0.0f);

    for (int kk = 0; kk < BATCH; kk += 4) {
        const int kl = kk + 2 * half;
        const v2f a = *(const v2f*)(arow + kl);              // global_load_b64
        const float* __restrict__ brow = Bv + (size_t)kl * NDIM + n0 + l15;
#pragma unroll
        for (int t = 0; t < NT; ++t) {
            v2f b;
            b.x = brow[t * 16];
            b.y = brow[t * 16 + NDIM];
            acc[t] = __builtin_amdgcn_wmma_f32_16x16x4_f32(
                /*neg_a=*/false, a, /*neg_b=*/false, b,
                /*c_mod=*/(short)0, acc[t],
                /*reuse_a=*/false, /*reuse_b=*/false);
        }
    }

    // Fused epilogue: bit-explode each C element to 32 pulse floats.
#pragma unroll
    for (int t = 0; t < NT; ++t) {
#pragma unroll
        for (int r = 0; r < 8; ++r) {
            const uint32_t u = __float_as_uint(acc[t][r]);
            const int k = tileM * 16 + r + 8 * half;
            const int n = n0 + t * 16 + l15;
            float* __restrict__ dst = out + ((size_t)k * NDIM + n) * 32;
#pragma unroll
            for (int q = 0; q < 8; ++q) {
                v4f v;
                v.x = (float)((u >> (31 - (q * 4 + 0))) & 1u);
                v.y = (float)((u >> (31 - (q * 4 + 1))) & 1u);
                v.z = (float)((u >> (31 - (q * 4 + 2))) & 1u);
                v.w = (float)((u >> (31 - (q * 4 + 3))) & 1u);
                *(v4f*)(dst + q * 4) = v;                    // global_store_b128
            }
        }
    }
}

extern "C" void kernel_launch(void* const* d_in, const int* in_sizes, int n_in,
                              void* d_out, int out_size, void* d_ws, size_t ws_size,
                              hipStream_t stream) {
    const float* A = (const float*)d_in[0];   // [512,1024,32] pulses
    const float* B = (const float*)d_in[1];   // [512,1024,32] pulses
    float* out = (float*)d_out;               // [1024,1024,32] pulses

    float* AvT = (float*)d_ws;                       // 1024*512 fp32 = 2 MB
    float* Bv  = AvT + (size_t)KDIM * BATCH;         // 512*1024 fp32 = 2 MB

    const int total = 2 * BATCH * KDIM;              // 1,048,576 values
    decode_kernel<<<total / 256, 256, 0, stream>>>(A, B, AvT, Bv);

    // 2048 waves (64 M-tiles x 32 N-strips), 8 waves/block -> 256 blocks.
    gemm_encode_kernel<<<256, 256, 0, stream>>>(AvT, Bv, out);
}